// ModelNew_25056839205197
// MI455X (gfx1250) — compile-verified
//
#include <hip/hip_runtime.h>
#include <math.h>

// Problem constants (from reference)
#define BB   16
#define CC   512
#define C2K  256
#define HH   64
#define WW   64
#define HWN  4096          // H*W
#define HW4  1024          // HW/4

typedef __attribute__((ext_vector_type(2))) float v2f;
typedef __attribute__((ext_vector_type(8))) float v8f;

static __device__ __forceinline__ v8f wmma_f32(v2f a, v2f b, v8f c) {
    // V_WMMA_F32_16X16X4_F32 : D = A(16x4) x B(4x16) + C(16x16), fp32 exact
    return __builtin_amdgcn_wmma_f32_16x16x4_f32(false, a, false, b, (short)0, c, false, false);
}

// ---------------------------------------------------------------------------
// K1: ch_wq[b,n] = sum_c wq_r[c] * x[b,c,n]       grid(16,16) x 256
// ---------------------------------------------------------------------------
__global__ void __launch_bounds__(256) k_chwq(const float* __restrict__ x,
                                              const float* __restrict__ wq,
                                              float* __restrict__ out) {
    const int b = blockIdx.y;
    const int n = blockIdx.x * 256 + threadIdx.x;
    const float* xb = x + (size_t)b * CC * HWN;
    float acc = 0.f;
    #pragma unroll 4
    for (int c = 0; c < CC; ++c)
        acc += wq[c] * xb[(size_t)c * HWN + n];
    out[(size_t)b * HWN + n] = acc;
}

// ---------------------------------------------------------------------------
// K2: p[b,:] = softmax over HW of ch_wq[b,:]      grid(16) x 256
// ---------------------------------------------------------------------------
__global__ void __launch_bounds__(256) k_softmax_hw(const float* __restrict__ z,
                                                    float* __restrict__ p) {
    const int b = blockIdx.x;
    const int tid = threadIdx.x;
    const float4* zr = (const float4*)(z + (size_t)b * HWN);
    float4 v[4];
    float mx = -1e30f;
    #pragma unroll
    for (int i = 0; i < 4; ++i) {
        v[i] = zr[tid + 256 * i];
        mx = fmaxf(mx, fmaxf(fmaxf(v[i].x, v[i].y), fmaxf(v[i].z, v[i].w)));
    }
    __shared__ float red[256];
    red[tid] = mx; __syncthreads();
    for (int s = 128; s > 0; s >>= 1) {
        if (tid < s) red[tid] = fmaxf(red[tid], red[tid + s]);
        __syncthreads();
    }
    mx = red[0]; __syncthreads();
    float sum = 0.f;
    #pragma unroll
    for (int i = 0; i < 4; ++i) {
        v[i].x = __expf(v[i].x - mx); v[i].y = __expf(v[i].y - mx);
        v[i].z = __expf(v[i].z - mx); v[i].w = __expf(v[i].w - mx);
        sum += v[i].x + v[i].y + v[i].z + v[i].w;
    }
    red[tid] = sum; __syncthreads();
    for (int s = 128; s > 0; s >>= 1) {
        if (tid < s) red[tid] += red[tid + s];
        __syncthreads();
    }
    const float inv = 1.f / red[0];
    float4* pr = (float4*)(p + (size_t)b * HWN);
    #pragma unroll
    for (int i = 0; i < 4; ++i) {
        float4 o; o.x = v[i].x * inv; o.y = v[i].y * inv;
        o.z = v[i].z * inv; o.w = v[i].w * inv;
        pr[tid + 256 * i] = o;
    }
}

// ---------------------------------------------------------------------------
// K3: xsum[b,c] = sum_n x[b,c,n];  xp[b,c] = sum_n x[b,c,n]*p[b,n]
//     grid(512,16) x 256
// ---------------------------------------------------------------------------
__global__ void __launch_bounds__(256) k_rowred(const float* __restrict__ x,
                                                const float* __restrict__ p,
                                                float* __restrict__ xsum,
                                                float* __restrict__ xp) {
    const int b = blockIdx.y, c = blockIdx.x, tid = threadIdx.x;
    const float4* xr = (const float4*)(x + ((size_t)b * CC + c) * HWN);
    const float4* pr = (const float4*)(p + (size_t)b * HWN);
    float as = 0.f, ap = 0.f;
    #pragma unroll
    for (int i = tid; i < HW4; i += 256) {
        float4 xv = xr[i], pv = pr[i];
        as += xv.x + xv.y + xv.z + xv.w;
        ap += xv.x * pv.x + xv.y * pv.y + xv.z * pv.z + xv.w * pv.w;
    }
    __shared__ float r1[256], r2[256];
    r1[tid] = as; r2[tid] = ap; __syncthreads();
    for (int s = 128; s > 0; s >>= 1) {
        if (tid < s) { r1[tid] += r1[tid + s]; r2[tid] += r2[tid + s]; }
        __syncthreads();
    }
    if (tid == 0) {
        xsum[b * CC + c] = r1[0];
        xp[b * CC + c]   = r2[0];
    }
}

// ---------------------------------------------------------------------------
// K4: fused small-GEMM chain on one workgroup (16 wave32 waves) using
//     V_WMMA_F32_16X16X4_F32.
//   d1: ch_wz[16,256]  = xp[16,512]  x wv_r^T
//   d2: chw[16,512]    = sigmoid(ch_wz x w_up^T)      (-> ch_weight)
//   d3: spq[16,256]    = (chw*xsum)[16,512] x wq_l^T / HW
//   d4: q[16,256]      = softmax over C2 (per-wave shfl reduction)
//   d5: g[16,512]      = (q x wv_l) * chw             (-> gate for wz pass)
// ---------------------------------------------------------------------------
__global__ void __launch_bounds__(512) k_fused_small(const float* __restrict__ xp,
                                                     const float* __restrict__ xsum,
                                                     const float* __restrict__ wv_r,
                                                     const float* __restrict__ w_up,
                                                     const float* __restrict__ wq_l,
                                                     const float* __restrict__ wv_l,
                                                     float* __restrict__ chw_out,
                                                     float* __restrict__ g_out) {
    __shared__ float s_chwz[BB * C2K];   // 16 KB : ch_wz, then sp_wq
    __shared__ float s_chw [BB * CC];    // 32 KB : sigmoid ch_weight
    __shared__ float s_sq  [BB * CC];    // 32 KB : s = chw*xsum, then q (first 16KB)

    const int tid   = threadIdx.x;
    const int wave  = tid >> 5;
    const int lane  = tid & 31;
    const int mrow  = lane & 15;     // A-row / B-col / D-col index within tile
    const int khalf = lane >> 4;     // K half (A/B), M half (C/D)

    // ---- d1: ch_wz = xp x wv_r^T  (M=16,N=256,K=512); wave -> one 16-col tile
    {
        const int n0 = wave * 16;
        v8f acc = {};
        for (int k0 = 0; k0 < CC; k0 += 4) {
            const int ka = k0 + 2 * khalf;
            v2f a, b;
            a.x = xp[mrow * CC + ka];            a.y = xp[mrow * CC + ka + 1];
            b.x = wv_r[(n0 + mrow) * CC + ka];   b.y = wv_r[(n0 + mrow) * CC + ka + 1];
            acc = wmma_f32(a, b, acc);
        }
        #pragma unroll
        for (int r = 0; r < 8; ++r)
            s_chwz[(r + 8 * khalf) * C2K + n0 + mrow] = acc[r];
    }
    __syncthreads();

    // ---- d2: chw = sigmoid(ch_wz x w_up^T)  (M=16,N=512,K=256)
    #pragma unroll
    for (int t = 0; t < 2; ++t) {
        const int n0 = (wave * 2 + t) * 16;
        v8f acc = {};
        for (int k0 = 0; k0 < C2K; k0 += 4) {
            const int ka = k0 + 2 * khalf;
            v2f a, b;
            a.x = s_chwz[mrow * C2K + ka];        a.y = s_chwz[mrow * C2K + ka + 1];
            b.x = w_up[(n0 + mrow) * C2K + ka];   b.y = w_up[(n0 + mrow) * C2K + ka + 1];
            acc = wmma_f32(a, b, acc);
        }
        #pragma unroll
        for (int r = 0; r < 8; ++r) {
            const int m = r + 8 * khalf, n = n0 + mrow;
            const float sg = 1.f / (1.f + __expf(-acc[r]));
            s_chw[m * CC + n] = sg;
            chw_out[m * CC + n] = sg;
        }
    }
    __syncthreads();

    // ---- s = chw * xsum (elementwise into LDS)
    for (int i = tid; i < BB * CC; i += 512)
        s_sq[i] = s_chw[i] * xsum[i];
    __syncthreads();

    // ---- d3: sp_wq = s x wq_l^T / HW  (M=16,N=256,K=512) -> overwrite s_chwz
    {
        const int n0 = wave * 16;
        v8f acc = {};
        for (int k0 = 0; k0 < CC; k0 += 4) {
            const int ka = k0 + 2 * khalf;
            v2f a, b;
            a.x = s_sq[mrow * CC + ka];           a.y = s_sq[mrow * CC + ka + 1];
            b.x = wq_l[(n0 + mrow) * CC + ka];    b.y = wq_l[(n0 + mrow) * CC + ka + 1];
            acc = wmma_f32(a, b, acc);
        }
        #pragma unroll
        for (int r = 0; r < 8; ++r)
            s_chwz[(r + 8 * khalf) * C2K + n0 + mrow] = acc[r] * (1.f / (float)HWN);
    }
    __syncthreads();

    // ---- d4: q = softmax over C2=256, one wave32 per batch row
    {
        const int b = wave;                  // 16 waves == 16 rows
        float vals[8];
        float mx = -1e30f;
        #pragma unroll
        for (int i = 0; i < 8; ++i) {
            vals[i] = s_chwz[b * C2K + lane + 32 * i];
            mx = fmaxf(mx, vals[i]);
        }
        #pragma unroll
        for (int off = 16; off >= 1; off >>= 1)
            mx = fmaxf(mx, __shfl_xor(mx, off, 32));
        float sum = 0.f;
        #pragma unroll
        for (int i = 0; i < 8; ++i) { vals[i] = __expf(vals[i] - mx); sum += vals[i]; }
        #pragma unroll
        for (int off = 16; off >= 1; off >>= 1)
            sum += __shfl_xor(sum, off, 32);
        const float inv = 1.f / sum;
        #pragma unroll
        for (int i = 0; i < 8; ++i)
            s_sq[b * C2K + lane + 32 * i] = vals[i] * inv;   // q into s_sq (16KB)
    }
    __syncthreads();

    // ---- d5: g = (q x wv_l) * chw  (M=16,N=512,K=256)
    #pragma unroll
    for (int t = 0; t < 2; ++t) {
        const int n0 = (wave * 2 + t) * 16;
        v8f acc = {};
        for (int k0 = 0; k0 < C2K; k0 += 4) {
            const int ka = k0 + 2 * khalf;
            v2f a, b;
            a.x = s_sq[mrow * C2K + ka];          a.y = s_sq[mrow * C2K + ka + 1];
            b.x = wv_l[ka * CC + n0 + mrow];      b.y = wv_l[(ka + 1) * CC + n0 + mrow];
            acc = wmma_f32(a, b, acc);
        }
        #pragma unroll
        for (int r = 0; r < 8; ++r) {
            const int m = r + 8 * khalf, n = n0 + mrow;
            g_out[m * CC + n] = acc[r] * s_chw[m * CC + n];
        }
    }
}

// ---------------------------------------------------------------------------
// K5: sig_wz[b,n] = sigmoid( sum_c g[b,c] * x[b,c,n] )   grid(16,16) x 256
// ---------------------------------------------------------------------------
__global__ void __launch_bounds__(256) k_wz(const float* __restrict__ x,
                                            const float* __restrict__ g,
                                            float* __restrict__ sig) {
    const int b = blockIdx.y;
    const int n = blockIdx.x * 256 + threadIdx.x;
    const float* xb = x + (size_t)b * CC * HWN;
    const float* gb = g + b * CC;
    float acc = 0.f;
    #pragma unroll 4
    for (int c = 0; c < CC; ++c)
        acc += gb[c] * xb[(size_t)c * HWN + n];
    sig[(size_t)b * HWN + n] = 1.f / (1.f + __expf(-acc));
}

// ---------------------------------------------------------------------------
// K6: out[b,c,n] = x[b,c,n] * chw[b,c] * sig_wz[b,n]     grid(32768) x 256
// ---------------------------------------------------------------------------
__global__ void __launch_bounds__(256) k_final(const float* __restrict__ x,
                                               const float* __restrict__ chw,
                                               const float* __restrict__ sig,
                                               float* __restrict__ out) {
    const size_t idx = (size_t)blockIdx.x * 256 + threadIdx.x;   // float4 units
    const int n4 = (int)(idx & (HW4 - 1));
    const int c  = (int)((idx >> 10) & (CC - 1));
    const int b  = (int)(idx >> 19);
    const float w = chw[b * CC + c];
    const float4 xv = ((const float4*)x)[idx];
    const float4 sv = ((const float4*)(sig + (size_t)b * HWN))[n4];
    float4 o;
    o.x = xv.x * w * sv.x; o.y = xv.y * w * sv.y;
    o.z = xv.z * w * sv.z; o.w = xv.w * w * sv.w;
    ((float4*)out)[idx] = o;
}

// ---------------------------------------------------------------------------
extern "C" void kernel_launch(void* const* d_in, const int* in_sizes, int n_in,
                              void* d_out, int out_size, void* d_ws, size_t ws_size,
                              hipStream_t stream) {
    const float* x    = (const float*)d_in[0];
    const float* wq_r = (const float*)d_in[1];
    const float* wv_r = (const float*)d_in[2];
    const float* w_up = (const float*)d_in[3];
    const float* wq_l = (const float*)d_in[4];
    const float* wv_l = (const float*)d_in[5];
    float* out = (float*)d_out;

    float* ws    = (float*)d_ws;
    float* chwq  = ws;                    // [B,HW]   65536
    float* p     = ws + 65536;            // [B,HW]   65536
    float* xsum  = ws + 131072;           // [B,C]     8192
    float* xp    = ws + 139264;           // [B,C]     8192
    float* chw   = ws + 147456;           // [B,C]     8192
    float* g     = ws + 155648;           // [B,C]     8192
    float* sig   = ws + 163840;           // [B,HW]   65536

    k_chwq      <<<dim3(HWN / 256, BB), 256, 0, stream>>>(x, wq_r, chwq);
    k_softmax_hw<<<BB, 256, 0, stream>>>(chwq, p);
    k_rowred    <<<dim3(CC, BB), 256, 0, stream>>>(x, p, xsum, xp);
    k_fused_small<<<1, 512, 0, stream>>>(xp, xsum, wv_r, w_up, wq_l, wv_l, chw, g);
    k_wz        <<<dim3(HWN / 256, BB), 256, 0, stream>>>(x, g, sig);
    k_final     <<<(BB * CC * HW4) / 256, 256, 0, stream>>>(x, chw, sig, out);
}